// CGRegressorAdapter_69320772158305
// MI455X (gfx1250) — compile-verified
//
#include <hip/hip_runtime.h>
#include <hip/hip_bf16.h>

#define Bz 32
#define Nn 2048
#define Ee 8192
#define Hh 128
#define NODES (Bz * Nn)          // 65536
#define NEDGES (Bz * Ee)         // 262144

typedef float v2f __attribute__((ext_vector_type(2)));
typedef float v8f __attribute__((ext_vector_type(8)));

__device__ __forceinline__ v8f wmma4(v2f a, v2f b, v8f c) {
  return __builtin_amdgcn_wmma_f32_16x16x4_f32(false, a, false, b, (short)0, c,
                                               false, false);
}

// ---------------------------------------------------------------------------
// x = embed_table[ind] + shapes @ shape_w     (one block per node, 128 thr)
// ---------------------------------------------------------------------------
__global__ __launch_bounds__(128) void embed_kernel(
    const int* __restrict__ inds, const float* __restrict__ shapes,
    const float* __restrict__ table, const float* __restrict__ shape_w,
    float* __restrict__ x)
{
  int node = blockIdx.x;
  int f = threadIdx.x;
  int id = inds[node];
  float s0 = shapes[node * 4 + 0];
  float s1 = shapes[node * 4 + 1];
  float s2 = shapes[node * 4 + 2];
  float s3 = shapes[node * 4 + 3];
  float v = table[(size_t)id * Hh + f];
  v += s0 * shape_w[0 * Hh + f];
  v += s1 * shape_w[1 * Hh + f];
  v += s2 * shape_w[2 * Hh + f];
  v += s3 * shape_w[3 * Hh + f];
  x[(size_t)node * Hh + f] = v;
}

// ---------------------------------------------------------------------------
// zero-fill (graph-capture-safe scratch init)
// ---------------------------------------------------------------------------
__global__ __launch_bounds__(256) void zero_kernel(float* __restrict__ p, size_t n)
{
  size_t i = (size_t)blockIdx.x * blockDim.x + threadIdx.x;
  if (i < n) p[i] = 0.0f;
}

// ---------------------------------------------------------------------------
// Repack one 128x128 row-major weight matrix per block into the WMMA B-fragment
// order: Wp[((kq*2 + kh)*128 + n)*2 + r] = W[kq*4 + 2*kh + r][n]
// so a lane's (b.x, b.y) = (W[k][n], W[k+1][n]) becomes one contiguous b64.
// ---------------------------------------------------------------------------
__global__ __launch_bounds__(256) void pack_w(const float* __restrict__ src,
                                              float* __restrict__ dst)
{
  size_t base = (size_t)blockIdx.x * (Hh * Hh);
  for (int idx = threadIdx.x; idx < Hh * Hh; idx += 256) {
    int k = idx >> 7, n = idx & 127;
    int kq = k >> 2, kh = (k >> 1) & 1, r = k & 1;
    dst[base + ((((kq * 2 + kh) << 7) + n) << 1) + r] = src[base + idx];
  }
}

// ---------------------------------------------------------------------------
// agg[dst] += h[src] over all edges.  One wave32 per edge, float4 per lane.
// agg (33.5 MB) is L2-resident -> atomics stay on-chip.
// ---------------------------------------------------------------------------
__global__ __launch_bounds__(256) void edge_scatter(
    const int* __restrict__ edge_index, const float* __restrict__ h,
    float* __restrict__ agg)
{
  int gwarp = (int)(((size_t)blockIdx.x * blockDim.x + threadIdx.x) >> 5);
  int lane = threadIdx.x & 31;
  if (gwarp >= NEDGES) return;
  int b = gwarp / Ee;
  int e = gwarp - b * Ee;
  const int* ei = edge_index + (size_t)b * 2 * Ee;
  int src = ei[e] + b * Nn;
  int dst = ei[Ee + e] + b * Nn;
  const float4 v = *(const float4*)(h + (size_t)src * Hh + lane * 4);
  float* ad = agg + (size_t)dst * Hh + lane * 4;
  unsafeAtomicAdd(ad + 0, v.x);
  unsafeAtomicAdd(ad + 1, v.y);
  unsafeAtomicAdd(ad + 2, v.z);
  unsafeAtomicAdd(ad + 3, v.w);
}

// ---------------------------------------------------------------------------
// out[M,128] = relu?( sum_{p<NPAIRS} A_p @ W_p + bias ),  K = 128 per pair.
// W_p must be in pack_w layout. One wave computes a 32x128 strip:
//   two A fragments (rows m0..m0+15, m0+16..m0+31), 8 N-tiles, 16 v8f acc.
//   Per k-step: 2 a b64 loads + 8 packed-B b64 loads feed 16 WMMAs.
// NPAIRS templated so the pointer arrays SROA away -> global_load (not flat).
// Grid sized exactly -> EXEC all-ones around WMMA.
// ---------------------------------------------------------------------------
template <int NPAIRS>
__global__ __launch_bounds__(128) void gconv_gemm(
    const float* __restrict__ A0, const float* __restrict__ P0,
    const float* __restrict__ A1, const float* __restrict__ P1,
    const float* __restrict__ A2, const float* __restrict__ P2,
    const float* __restrict__ A3, const float* __restrict__ P3,
    const float* __restrict__ bias, float* __restrict__ out, int do_relu)
{
  const int wave = threadIdx.x >> 5;
  const int lane = threadIdx.x & 31;
  const int m0 = (blockIdx.x * 4 + wave) * 32;   // 32 rows per wave
  const int mrow = lane & 15;                    // M row in tile / N col for B,C,D
  const int kh = lane >> 4;                      // K half-select

  v8f acc0[8], acc1[8];
#pragma unroll
  for (int i = 0; i < 8; ++i) {
    acc0[i] = v8f{0.f, 0.f, 0.f, 0.f, 0.f, 0.f, 0.f, 0.f};
    acc1[i] = v8f{0.f, 0.f, 0.f, 0.f, 0.f, 0.f, 0.f, 0.f};
  }

  const float* As[4] = {A0, A1, A2, A3};
  const float* Ps[4] = {P0, P1, P2, P3};

#pragma unroll
  for (int p = 0; p < NPAIRS; ++p) {
    const float* __restrict__ a0row = As[p] + (size_t)(m0 + mrow) * Hh;
    const float* __restrict__ a1row = a0row + 16 * Hh;
    const v2f* __restrict__ Pp = (const v2f*)Ps[p];
    for (int kq = 0; kq < 32; ++kq) {
      const int kbase = kq * 4 + 2 * kh;
      v2f a0 = *(const v2f*)(a0row + kbase);
      v2f a1 = *(const v2f*)(a1row + kbase);
      const v2f* __restrict__ brow = Pp + (((kq * 2 + kh) << 7) + mrow);
#pragma unroll
      for (int nt = 0; nt < 8; ++nt) {
        v2f b = brow[nt * 16];
        acc0[nt] = wmma4(a0, b, acc0[nt]);
        acc1[nt] = wmma4(a1, b, acc1[nt]);
      }
    }
  }

  // epilogue: bias + optional relu, store per documented C/D layout
#pragma unroll
  for (int nt = 0; nt < 8; ++nt) {
    const int col = nt * 16 + mrow;
    const float bv = bias[col];
#pragma unroll
    for (int j = 0; j < 8; ++j) {
      const int row = m0 + j + 8 * kh;
      float v0 = acc0[nt][j] + bv;
      float v1 = acc1[nt][j] + bv;
      if (do_relu) { v0 = fmaxf(v0, 0.0f); v1 = fmaxf(v1, 0.0f); }
      out[(size_t)row * Hh + col] = v0;
      out[(size_t)(row + 16) * Hh + col] = v1;
    }
  }
}

// ---------------------------------------------------------------------------
// g[b,:] = concat(lat4[last], curr[last])   (grid 32, block 256)
// ---------------------------------------------------------------------------
__global__ __launch_bounds__(256) void gather_kernel(
    const float* __restrict__ lat4, const float* __restrict__ curr,
    const int* __restrict__ last_idx, float* __restrict__ g)
{
  int b = blockIdx.x;
  int t = threadIdx.x;
  size_t node = (size_t)b * Nn + last_idx[b];
  g[(size_t)b * 256 + t] =
      (t < Hh) ? lat4[node * Hh + t] : curr[node * Hh + (t - Hh)];
}

// ---------------------------------------------------------------------------
// tiny head GEMM: out[b,n] = relu?( sum_k in[b,k]*W[k,n] + bias[n] )
// ---------------------------------------------------------------------------
__global__ __launch_bounds__(128) void head_gemm(
    const float* __restrict__ in, const float* __restrict__ W,
    const float* __restrict__ bias, float* __restrict__ out,
    int K, int Nout, int do_relu)
{
  int b = blockIdx.x;
  int n = threadIdx.x;
  if (n >= Nout) return;
  float acc = bias[n];
  for (int k = 0; k < K; ++k) acc += in[(size_t)b * K + k] * W[(size_t)k * Nout + n];
  if (do_relu) acc = fmaxf(acc, 0.0f);
  out[(size_t)b * Nout + n] = acc;
}

// ---------------------------------------------------------------------------
extern "C" void kernel_launch(void* const* d_in, const int* in_sizes, int n_in,
                              void* d_out, int out_size, void* d_ws, size_t ws_size,
                              hipStream_t stream) {
  const int*   inds    = (const int*)d_in[0];
  const float* shapes  = (const float*)d_in[1];
  const int*   edge    = (const int*)d_in[2];
  const int*   last    = (const int*)d_in[3];
  const float* table   = (const float*)d_in[4];
  const float* shape_w = (const float*)d_in[5];
  const float* bWself  = (const float*)d_in[6];   // [4,128,128]
  const float* bWnbr   = (const float*)d_in[7];
  const float* bb      = (const float*)d_in[8];   // [4,128]
  const float* aWself  = (const float*)d_in[9];   // [4,256,128] = 8 x 128x128
  const float* aWnbr   = (const float*)d_in[10];
  const float* ab      = (const float*)d_in[11];
  const float* hW1     = (const float*)d_in[12];  // [256,128]
  const float* hb1     = (const float*)d_in[13];
  const float* hWmid   = (const float*)d_in[14];  // [3,128,128]
  const float* hbmid   = (const float*)d_in[15];  // [3,128]
  const float* hW5     = (const float*)d_in[16];  // [128,1]
  const float* hb5     = (const float*)d_in[17];
  float* out = (float*)d_out;

  float* ws = (float*)d_ws;
  const size_t NF = (size_t)NODES * Hh;           // 8.39M floats per buffer
  const size_t WM = (size_t)Hh * Hh;              // 16384 floats per matrix
  float* lat[5];
  for (int i = 0; i < 5; ++i) lat[i] = ws + (size_t)i * NF;
  float* cbuf0 = ws + 5 * NF;
  float* cbuf1 = ws + 6 * NF;
  float* aggA  = ws + 7 * NF;
  float* aggB  = ws + 8 * NF;
  float* g   = ws + 9 * NF;        // 32*256
  float* y1  = g + Bz * 256;       // 32*128
  float* y2  = y1 + Bz * Hh;       // 32*128
  float* pBS = y2 + Bz * Hh;       // packed base_Wself   (4 matrices)
  float* pBN = pBS + 4 * WM;       // packed base_Wnbr    (4 matrices)
  float* pAS = pBN + 4 * WM;       // packed adapt_Wself  (8 matrices)
  float* pAN = pAS + 8 * WM;       // packed adapt_Wnbr   (8 matrices)

  const int zgrid = (int)((NF + 255) / 256);
  const int sgrid = (NEDGES * 32) / 256;          // one wave per edge
  const int ggrid = NODES / 128;                  // 4 waves/block * 32 rows

  // one-time weight repack into WMMA B-fragment order (1.5 MB total)
  pack_w<<<4, 256, 0, stream>>>(bWself, pBS);
  pack_w<<<4, 256, 0, stream>>>(bWnbr,  pBN);
  pack_w<<<8, 256, 0, stream>>>(aWself, pAS);
  pack_w<<<8, 256, 0, stream>>>(aWnbr,  pAN);

  // embed
  embed_kernel<<<NODES, 128, 0, stream>>>(inds, shapes, table, shape_w, lat[0]);

  // base encoder: lat[i+1] = relu(lat[i]@Wself + agg@Wnbr + b)
  for (int i = 0; i < 4; ++i) {
    zero_kernel<<<zgrid, 256, 0, stream>>>(aggA, NF);
    edge_scatter<<<sgrid, 256, 0, stream>>>(edge, lat[i], aggA);
    gconv_gemm<2><<<ggrid, 128, 0, stream>>>(
        lat[i], pBS + (size_t)i * WM,
        aggA,   pBN + (size_t)i * WM,
        nullptr, nullptr, nullptr, nullptr,
        bb + (size_t)i * Hh, lat[i + 1], 1);
  }

  // adapter: inp = concat(lat[i+1], curr) decomposed into K=128 halves
  const float* curr = lat[0];
  float* nxt = cbuf0;
  for (int i = 0; i < 4; ++i) {
    zero_kernel<<<zgrid, 256, 0, stream>>>(aggA, NF);
    zero_kernel<<<zgrid, 256, 0, stream>>>(aggB, NF);
    edge_scatter<<<sgrid, 256, 0, stream>>>(edge, lat[i + 1], aggA);
    edge_scatter<<<sgrid, 256, 0, stream>>>(edge, curr, aggB);
    gconv_gemm<4><<<ggrid, 128, 0, stream>>>(
        lat[i + 1], pAS + (size_t)(2 * i) * WM,       // top half of Wself
        curr,       pAS + (size_t)(2 * i + 1) * WM,   // bottom half of Wself
        aggA,       pAN + (size_t)(2 * i) * WM,       // top half of Wnbr
        aggB,       pAN + (size_t)(2 * i + 1) * WM,   // bottom half of Wnbr
        ab + (size_t)i * Hh, nxt, 1);
    curr = nxt;
    nxt = (i & 1) ? cbuf0 : cbuf1;
  }

  // aggregator + regression head
  gather_kernel<<<Bz, 256, 0, stream>>>(lat[4], curr, last, g);
  head_gemm<<<Bz, 128, 0, stream>>>(g,  hW1,                 hb1,            y1, 256, Hh, 0);
  head_gemm<<<Bz, 128, 0, stream>>>(y1, hWmid,               hbmid,          y2, Hh,  Hh, 1);
  head_gemm<<<Bz, 128, 0, stream>>>(y2, hWmid + Hh * Hh,     hbmid + Hh,     y1, Hh,  Hh, 0);
  head_gemm<<<Bz, 128, 0, stream>>>(y1, hWmid + 2 * Hh * Hh, hbmid + 2 * Hh, y2, Hh,  Hh, 1);
  head_gemm<<<Bz, 128, 0, stream>>>(y2, hW5,                 hb5,            out, Hh, 1, 0);
}